// InstanceRouteOptimizationArea_53558242181774
// MI455X (gfx1250) — compile-verified
//
#include <hip/hip_runtime.h>

// ---------------------------------------------------------------------------
// InstanceRouteOptimizationArea for MI455X (gfx1250)
//
// Phase 1: per-net bbox + h/v density               (gather, trivial BW)
// Phase 2: util map = two 256x100000 @ 100000x256 f32 GEMMs, computed with
//          V_WMMA_F32_16X16X4_F32 (f32 accumulate -> matches fp32 reference),
//          operands generated on the fly from LDS-staged (AoS) net records.
//          Uniform compile-time trip counts keep EXEC all-ones around WMMA.
// Phase 3: per-instance <=3x3-bin overlap dot with util map (L2 resident).
// ---------------------------------------------------------------------------

#define NBX 256
#define NBY 256
constexpr float BSX = 1000.0f / 256.0f;   // 3.90625
constexpr float BSY = 1000.0f / 256.0f;
constexpr float INV_H = 1.0f / (BSX * BSY * 1.5625f);
constexpr float INV_V = 1.0f / (BSX * BSY * 1.45f);
constexpr float MIN_RATE = 0.5f;
constexpr float MAX_RATE = 2.0f;

typedef __attribute__((ext_vector_type(2))) float v2f;
typedef __attribute__((ext_vector_type(8))) float v8f;

// ---------------- Phase 1: per-net bbox + densities ----------------
__global__ void net_bbox_kernel(const float* __restrict__ pin_pos,
                                const float* __restrict__ net_weights,
                                const int*   __restrict__ netpin_start,
                                const int*   __restrict__ flat_netpin,
                                int num_pins, int num_nets,
                                float* __restrict__ o_xmin, float* __restrict__ o_xmax,
                                float* __restrict__ o_ymin, float* __restrict__ o_ymax,
                                float* __restrict__ o_hd,   float* __restrict__ o_vd) {
    int n = blockIdx.x * blockDim.x + threadIdx.x;
    if (n >= num_nets) return;
    int s = netpin_start[n];
    int e = netpin_start[n + 1];
    float xmn = 3.0e38f, xmx = -3.0e38f, ymn = 3.0e38f, ymx = -3.0e38f;
    for (int j = s; j < e; ++j) {
        int p = flat_netpin[j];
        float px = pin_pos[p];
        float py = pin_pos[p + num_pins];
        xmn = fminf(xmn, px); xmx = fmaxf(xmx, px);
        ymn = fminf(ymn, py); ymx = fmaxf(ymx, py);
    }
    float wx = xmx - xmn;
    float wy = ymx - ymn;
    float area = fmaxf(wx * wy, 1e-6f);
    float w = net_weights[n];
    o_xmin[n] = xmn; o_xmax[n] = xmx;
    o_ymin[n] = ymn; o_ymax[n] = ymx;
    o_hd[n] = w * wx / area;
    o_vd[n] = w * wy / area;
}

// ---------------- Phase 2: util map via f32 WMMA GEMM ----------------
// Grid: 256 workgroups, one 16x16 output tile each (tx = blk>>4, ty = blk&15).
// Block: 256 threads = 8 wave32s. K dim = nets, staged in LDS in chunks.
#define CHUNK   1024
#define GROUPS  (CHUNK / 4)        // 4 nets (K=4) per WMMA
#define GPW     (GROUPS / 8)       // groups per wave per chunk (uniform)

__global__ __launch_bounds__(256) void util_map_wmma_kernel(
        const float* __restrict__ g_xmin, const float* __restrict__ g_xmax,
        const float* __restrict__ g_ymin, const float* __restrict__ g_ymax,
        const float* __restrict__ g_hd,   const float* __restrict__ g_vd,
        int num_nets, float* __restrict__ util) {
    // AoS: 8 floats per net record {xmin,xmax,ymin,ymax,hd,vd,0,0} -> 32 KB.
    // First 4096 floats reused afterwards as the cross-wave reduce buffer.
    __shared__ __align__(16) float smem[CHUNK * 8];

    const int tid  = threadIdx.x;
    const int wave = tid >> 5;
    const int lane = tid & 31;
    const int hf   = lane >> 4;                  // which K-pair this lane owns
    const int l    = lane & 15;                  // M index (A) / N index (B)
    const int tx   = blockIdx.x >> 4;
    const int ty   = blockIdx.x & 15;

    // Loop-invariant bin edges for this lane's row (x) / column (y)
    const float bxlo = (float)(tx * 16 + l) * BSX;
    const float bxhi = bxlo + BSX;
    const float bylo = (float)(ty * 16 + l) * BSY;
    const float byhi = bylo + BSY;

    v8f ch = {};   // H-map accumulator tile
    v8f cv = {};   // V-map accumulator tile

    for (int k0 = 0; k0 < num_nets; k0 += CHUNK) {
        __syncthreads();
        for (int j = tid; j < CHUNK; j += 256) {
            int n = k0 + j;
            bool ok = (n < num_nets);
            float4 r0, r1;
            r0.x = ok ? g_xmin[n] : 1.0f;        // empty interval -> overlap 0
            r0.y = ok ? g_xmax[n] : 0.0f;
            r0.z = ok ? g_ymin[n] : 1.0f;
            r0.w = ok ? g_ymax[n] : 0.0f;
            r1.x = ok ? g_hd[n]   : 0.0f;
            r1.y = ok ? g_vd[n]   : 0.0f;
            r1.z = 0.0f; r1.w = 0.0f;
            ((float4*)&smem[j * 8])[0] = r0;
            ((float4*)&smem[j * 8])[1] = r1;
            // Warm L0/L2 for the next chunk (global_prefetch_b8, no counter).
            int np = n + CHUNK;
            if (np < num_nets) {
                __builtin_prefetch(&g_xmin[np], 0, 1);
                __builtin_prefetch(&g_xmax[np], 0, 1);
                __builtin_prefetch(&g_ymin[np], 0, 1);
                __builtin_prefetch(&g_ymax[np], 0, 1);
                __builtin_prefetch(&g_hd[np],   0, 1);
                __builtin_prefetch(&g_vd[np],   0, 1);
            }
        }
        __syncthreads();

        // Uniform trip count: no exec-masked looping around WMMA.
#pragma unroll 4
        for (int gg = 0; gg < GPW; ++gg) {
            int g  = gg * 8 + wave;
            int p0 = g * 4 + hf * 2;             // this lane's two nets
            const float4* rec0 = (const float4*)&smem[p0 * 8];
            float4 q0 = rec0[0];                 // xmin,xmax,ymin,ymax
            float4 d0 = rec0[1];                 // hd, vd, -, -
            float4 q1 = rec0[2];                 // net p0+1
            float4 d1 = rec0[3];

            float ox0 = fmaxf(0.0f, fminf(q0.y, bxhi) - fmaxf(q0.x, bxlo));
            float ox1 = fmaxf(0.0f, fminf(q1.y, bxhi) - fmaxf(q1.x, bxlo));
            float oy0 = fmaxf(0.0f, fminf(q0.w, byhi) - fmaxf(q0.z, bylo));
            float oy1 = fmaxf(0.0f, fminf(q1.w, byhi) - fmaxf(q1.z, bylo));

            v2f ah; ah[0] = ox0 * d0.x; ah[1] = ox1 * d1.x;   // A for H map
            v2f av; av[0] = ox0 * d0.y; av[1] = ox1 * d1.y;   // A for V map
            v2f b;  b[0]  = oy0;        b[1]  = oy1;          // shared B

            ch = __builtin_amdgcn_wmma_f32_16x16x4_f32(
                     false, ah, false, b, (short)0, ch, false, false);
            cv = __builtin_amdgcn_wmma_f32_16x16x4_f32(
                     false, av, false, b, (short)0, cv, false, false);
        }
    }

    // Cross-wave reduction of the 16x16 partial tiles through LDS.
    __syncthreads();
#pragma unroll
    for (int v = 0; v < 8; ++v) {
        smem[wave * 256 + v * 32 + lane]        = ch[v];
        smem[2048 + wave * 256 + v * 32 + lane] = cv[v];
    }
    __syncthreads();

    float H = 0.0f, V = 0.0f;
#pragma unroll
    for (int w = 0; w < 8; ++w) {
        H += smem[w * 256 + tid];
        V += smem[2048 + w * 256 + tid];
    }
    // C/D layout: vgpr v, lanes 0-15 -> M=v, lanes 16-31 -> M=v+8; N=lane&15
    int v  = tid >> 5;
    int ln = tid & 31;
    int m  = v + 8 * (ln >> 4);
    int n  = ln & 15;
    float hu = H * INV_H;
    float vu = V * INV_V;
    float u  = fminf(fmaxf(fmaxf(hu, vu), MIN_RATE), MAX_RATE);
    util[(tx * 16 + m) * NBY + (ty * 16 + n)] = u;
}

// ---------------- Phase 3: per-instance route area ----------------
__global__ void instance_area_kernel(const float* __restrict__ pos,
                                     const float* __restrict__ nszx,
                                     const float* __restrict__ nszy,
                                     const float* __restrict__ util,
                                     int num_nodes, int num_movable,
                                     float* __restrict__ out) {
    int i = blockIdx.x * blockDim.x + threadIdx.x;
    if (i >= num_movable) return;
    float xlo = pos[i];
    float ylo = pos[num_nodes + i];
    float xhi = xlo + nszx[i];
    float yhi = ylo + nszy[i];

    int bx0 = max(0, min(NBX - 1, (int)floorf(xlo / BSX)));
    int bx1 = max(0, min(NBX - 1, (int)floorf(xhi / BSX)));
    int by0 = max(0, min(NBY - 1, (int)floorf(ylo / BSY)));
    int by1 = max(0, min(NBY - 1, (int)floorf(yhi / BSY)));

    float acc = 0.0f;
    for (int bx = bx0; bx <= bx1; ++bx) {
        float blo = (float)bx * BSX;
        float ox  = fmaxf(0.0f, fminf(xhi, blo + BSX) - fmaxf(xlo, blo));
        float inner = 0.0f;
        for (int by = by0; by <= by1; ++by) {
            float bl = (float)by * BSY;
            float oy = fmaxf(0.0f, fminf(yhi, bl + BSY) - fmaxf(ylo, bl));
            inner = fmaf(util[bx * NBY + by], oy, inner);
        }
        acc = fmaf(ox, inner, acc);
    }
    out[i] = acc;
}

// ---------------- launch ----------------
extern "C" void kernel_launch(void* const* d_in, const int* in_sizes, int n_in,
                              void* d_out, int out_size, void* d_ws, size_t ws_size,
                              hipStream_t stream) {
    const float* pos          = (const float*)d_in[0];
    const float* pin_pos      = (const float*)d_in[1];
    const float* node_size_x  = (const float*)d_in[2];
    const float* node_size_y  = (const float*)d_in[3];
    const float* net_weights  = (const float*)d_in[4];
    const int*   netpin_start = (const int*)d_in[5];
    const int*   flat_netpin  = (const int*)d_in[6];

    const int num_nets    = in_sizes[5] - 1;   // 100000
    const int num_pins    = in_sizes[1] / 2;   // 400000
    const int num_nodes   = in_sizes[2];       // 120000
    const int num_movable = out_size;          // 100000

    float* ws    = (float*)d_ws;
    float* xmin  = ws;
    float* xmax  = ws + (size_t)num_nets;
    float* ymin  = ws + (size_t)num_nets * 2;
    float* ymax  = ws + (size_t)num_nets * 3;
    float* hd    = ws + (size_t)num_nets * 4;
    float* vd    = ws + (size_t)num_nets * 5;
    float* util  = ws + (size_t)num_nets * 6;  // 256*256 floats

    int blocks_net = (num_nets + 255) / 256;
    net_bbox_kernel<<<blocks_net, 256, 0, stream>>>(
        pin_pos, net_weights, netpin_start, flat_netpin,
        num_pins, num_nets, xmin, xmax, ymin, ymax, hd, vd);

    util_map_wmma_kernel<<<256, 256, 0, stream>>>(
        xmin, xmax, ymin, ymax, hd, vd, num_nets, util);

    int blocks_node = (num_movable + 255) / 256;
    instance_area_kernel<<<blocks_node, 256, 0, stream>>>(
        pos, node_size_x, node_size_y, util,
        num_nodes, num_movable, (float*)d_out);
}